// BERTPTBaseline_35562329211831
// MI455X (gfx1250) — compile-verified
//
#include <hip/hip_runtime.h>

typedef __attribute__((ext_vector_type(2))) float v2f;
typedef __attribute__((ext_vector_type(8))) float v8f;

#define BDIM 64
#define SDIM 512
#define HDIM 1024
#define TDIM 3

// workspace (float) layout
#define WS_SENT 0    // 64*3 per-batch sentiment dot sums
#define WS_MASK 192
#define WS_ASP  193
#define WS_OPI  194

// ---------------------------------------------------------------------------
// zero the accumulators in d_ws (harness poisons it with 0xAA)
__global__ void init_ws_kernel(float* __restrict__ ws) {
    int i = threadIdx.x;
    if (i < 256) ws[i] = 0.0f;
}

// ---------------------------------------------------------------------------
// Fused single-pass kernel: for each 16-row tile of hidden (B*S rows, H=1024),
// compute a 16x16 f32 WMMA GEMM against a zero-padded 16-column weight tile:
//   cols 0-2  : W_aspect   -> aspect_logits (+bias)
//   cols 3-5  : W_opinion  -> opinion_logits (+bias)
//   cols 6-8  : W_sent     -> per-batch atomic sums (for pooled sentiment)
//   cols 9-15 : zero padding
// 4 waves / block, 16 rows / wave, K consumed in chunks of 512 staged in LDS.
__global__ __launch_bounds__(128) void fused_logits_kernel(
    const float* __restrict__ hidden,
    const float* __restrict__ Wa, const float* __restrict__ ba,
    const float* __restrict__ Wo, const float* __restrict__ bo,
    const float* __restrict__ Ws,
    float* __restrict__ aspect_out, float* __restrict__ opinion_out,
    float* __restrict__ sent_accum)
{
    __shared__ float lds_w[512 * 16];       // 32 KB K-chunk of padded weights

    const int tid  = threadIdx.x;
    const int wave = tid >> 5;
    const int lane = tid & 31;
    const int col  = lane & 15;             // B/C/D column (N)
    const int koff = (lane >> 4) << 1;      // A/B fragment: lanes 16-31 hold K+2,K+3

    const int rowTile = blockIdx.x * 64 + wave * 16;   // row = b*SDIM + s
    const int row     = rowTile + (lane & 15);         // A-fragment row (M)
    const int batch   = rowTile >> 9;                  // SDIM = 512, tile within batch

    v8f acc = {};   // 16x16 f32 accumulator (8 VGPRs)

    for (int kc = 0; kc < HDIM; kc += 512) {
        // --- stage padded weight chunk [kc, kc+512) x 16 into LDS ---
        for (int i = tid; i < 512 * 16; i += 128) {
            const int k = kc + (i >> 4);
            const int n = i & 15;
            float w = 0.0f;
            if (n < 3)      w = Wa[k * 3 + n];
            else if (n < 6) w = Wo[k * 3 + (n - 3)];
            else if (n < 9) w = Ws[k * 3 + (n - 6)];
            lds_w[i] = w;
        }
        __syncthreads();

        const float* arow = hidden + (size_t)row * HDIM + kc + koff;

        #pragma unroll 4
        for (int k0 = 0; k0 < 512; k0 += 4) {
            // A fragment: 16x4 f32, one 8-byte load per lane (streams hidden once)
            v2f a = *(const v2f*)(arow + k0);
            // B fragment: 4x16 f32 from LDS (lane = column, koff selects K pair)
            const int kb = (k0 + koff) << 4;
            v2f b;
            b.x = lds_w[kb + col];
            b.y = lds_w[kb + 16 + col];
            // D = A x B + C   (v_wmma_f32_16x16x4_f32)
            acc = __builtin_amdgcn_wmma_f32_16x16x4_f32(
                false, a, false, b, (short)0, acc, false, false);
        }
        __syncthreads();
    }

    // --- scatter results: C/D layout -> lane=N, vgpr r = M (+8 for lanes>=16)
    const int mbase = (lane >> 4) * 8;
    if (col < 3) {
        const float bias = ba[col];
        #pragma unroll
        for (int r = 0; r < 8; ++r)
            aspect_out[(size_t)(rowTile + mbase + r) * 3 + col] = acc[r] + bias;
    } else if (col < 6) {
        const float bias = bo[col - 3];
        #pragma unroll
        for (int r = 0; r < 8; ++r)
            opinion_out[(size_t)(rowTile + mbase + r) * 3 + (col - 3)] = acc[r] + bias;
    } else if (col < 9) {
        float s = 0.0f;
        #pragma unroll
        for (int r = 0; r < 8; ++r) s += acc[r];   // sum over this wave's 8 rows
        atomicAdd(&sent_accum[batch * 3 + (col - 6)], s);
    }
}

// ---------------------------------------------------------------------------
// CRF emit + transition scores and mask sum (wave32 shuffle reduction).
__global__ __launch_bounds__(256) void crf_kernel(
    const float* __restrict__ aspect_logits, const float* __restrict__ opinion_logits,
    const int* __restrict__ mask,
    const int* __restrict__ alab, const int* __restrict__ olab,
    const float* __restrict__ trans_a, const float* __restrict__ trans_o,
    float* __restrict__ ws)
{
    const int g = blockIdx.x * blockDim.x + threadIdx.x;  // g in [0, B*S)
    const int s = g & (SDIM - 1);

    const float mf = (float)mask[g];
    const int la = alab[g], lo = olab[g];

    float msum   = mf;
    float ascore = mf * aspect_logits[g * 3 + la];
    float oscore = mf * opinion_logits[g * 3 + lo];
    if (s > 0) {
        const float pm = mf * (float)mask[g - 1];
        ascore += pm * trans_a[alab[g - 1] * 3 + la];
        oscore += pm * trans_o[olab[g - 1] * 3 + lo];
    }

    // wave32 reduction
    for (int off = 16; off > 0; off >>= 1) {
        msum   += __shfl_down(msum, off);
        ascore += __shfl_down(ascore, off);
        oscore += __shfl_down(oscore, off);
    }
    if ((threadIdx.x & 31) == 0) {
        atomicAdd(&ws[WS_MASK], msum);
        atomicAdd(&ws[WS_ASP],  ascore);
        atomicAdd(&ws[WS_OPI],  oscore);
    }
}

// ---------------------------------------------------------------------------
// sentiment_logits = sent_accum/S + bias ; loss = -(asp+opi scores)/mask_sum
__global__ void finalize_kernel(const float* __restrict__ ws,
                                const float* __restrict__ bs,
                                float* __restrict__ sent_out,
                                float* __restrict__ loss_out)
{
    const int i = threadIdx.x;
    if (i < BDIM * TDIM)
        sent_out[i] = ws[WS_SENT + i] * (1.0f / (float)SDIM) + bs[i % 3];
    if (i == 0) {
        const float ms = ws[WS_MASK];
        loss_out[0] = -(ws[WS_ASP] / ms) - (ws[WS_OPI] / ms);
    }
}

// ---------------------------------------------------------------------------
extern "C" void kernel_launch(void* const* d_in, const int* in_sizes, int n_in,
                              void* d_out, int out_size, void* d_ws, size_t ws_size,
                              hipStream_t stream)
{
    (void)in_sizes; (void)n_in; (void)out_size; (void)ws_size;

    const float* hidden = (const float*)d_in[0];
    const int*   mask   = (const int*)  d_in[1];
    const int*   alab   = (const int*)  d_in[2];
    const int*   olab   = (const int*)  d_in[3];
    const float* Wa     = (const float*)d_in[4];
    const float* ba     = (const float*)d_in[5];
    const float* Wo     = (const float*)d_in[6];
    const float* bo     = (const float*)d_in[7];
    const float* Ws     = (const float*)d_in[8];
    const float* bs     = (const float*)d_in[9];
    const float* ta     = (const float*)d_in[10];
    const float* to     = (const float*)d_in[11];

    float* aspect_out  = (float*)d_out;
    float* opinion_out = aspect_out + (size_t)BDIM * SDIM * TDIM;
    float* sent_out    = opinion_out + (size_t)BDIM * SDIM * TDIM;
    float* loss_out    = sent_out + BDIM * TDIM;
    float* ws          = (float*)d_ws;

    init_ws_kernel<<<1, 256, 0, stream>>>(ws);

    // (B*S)/(16 rows * 4 waves) = 32768/64 = 512 blocks
    fused_logits_kernel<<<512, 128, 0, stream>>>(
        hidden, Wa, ba, Wo, bo, Ws, aspect_out, opinion_out, ws + WS_SENT);

    crf_kernel<<<(BDIM * SDIM) / 256, 256, 0, stream>>>(
        aspect_out, opinion_out, mask, alab, olab, ta, to, ws);

    finalize_kernel<<<1, 256, 0, stream>>>(ws, bs, sent_out, loss_out);
}